// Ultrasound_Rendering_76141180223717
// MI455X (gfx1250) — compile-verified
//
#include <hip/hip_runtime.h>
#include <hip/hip_bf16.h>

typedef __attribute__((ext_vector_type(2))) float v2f;
typedef __attribute__((ext_vector_type(8))) float v8f;

#define UH 2048
#define UW 2048

// PSF constants (fp32, matching the reference)
#define WAVELEN   3.08e-4f               // 1540/5e6
#define TONE_LEN  6.16e-4f               // 2*WAVELEN
#define HALF_AX   3.08e-4f               // TONE_LEN/2
#define HALF_LAT  6.16e-4f               // 1.0*WAVELEN*2
#define SW_PIX    (0.04f / 2048.0f)
#define SH_PIX    (0.05f / 2048.0f)
#define N_AX 12
#define N_LAT 31
#define AX_TAPS 23                        // 2*N_AX-1
#define LAT_TAPS 61                       // 2*N_LAT-1
#define AX_PAD 11
#define LAT_PAD 30
#define PI_F 3.14159265358979323846f

// conv tiling
#define OY 64
#define OX 64
#define IN_ROWS 96                        // OY + 22, rounded to 16
#define IN_COLS 124                       // OX + 60
#define IN_STRIDE 125
#define LAT_STRIDE 65

// ---------------------------------------------------------------------------
// Kernel 1: jittered PSF taps + normalization (trivial; one lane does it all)
// ---------------------------------------------------------------------------
__global__ void us_psf_weights(const float* __restrict__ rand_ax,
                               const float* __restrict__ rand_lat,
                               float* __restrict__ wsw) {
    if (threadIdx.x != 0) return;
    float ra = rand_ax[0], rl = rand_lat[0];
    float axv[AX_TAPS], latv[LAT_TAPS];
    float sa = 0.0f, sl = 0.0f;
    for (int i = 0; i < AX_TAPS; ++i) {
        float axc = (float)(i - (N_AX - 1)) * (HALF_AX / (float)(N_AX - 1)) + ra * SH_PIX;
        float t = 3.0f * (2.0f / TONE_LEN) * axc;
        axv[i] = expf(-0.5f * t * t);
        sa += axv[i];
    }
    for (int k = 0; k < LAT_TAPS; ++k) {
        float lc = (float)(k - (N_LAT - 1)) * (HALF_LAT / (float)(N_LAT - 1)) + rl * SW_PIX;
        float x = lc / WAVELEN;           // F_NUMBER*WAVELENGTH, F=1
        float s = (x == 0.0f) ? 1.0f : sinf(PI_F * x) / (PI_F * x);
        latv[k] = s * s;
        sl += latv[k];
    }
    float inv = 1.0f / (sa * sl);
    for (int i = 0; i < AX_TAPS; ++i)  wsw[i] = axv[i];
    for (int k = 0; k < LAT_TAPS; ++k) wsw[32 + k] = latv[k] * inv;
}

// ---------------------------------------------------------------------------
// Kernel 2: scatterer = exp(raw[...,0]).T  (+ scatterers_map = 20*log10(s+1))
// ---------------------------------------------------------------------------
__global__ void us_transpose_exp(const float4* __restrict__ raw4,
                                 float* __restrict__ scat,
                                 float* __restrict__ smap) {
    __shared__ float tile[32][33];
    int tx = threadIdx.x, ty = threadIdx.y;
    int x0 = blockIdx.x * 32, y0 = blockIdx.y * 32;
#pragma unroll
    for (int q = 0; q < 4; ++q) {
        int xl = ty + 8 * q;
        float v = raw4[(size_t)(x0 + xl) * UH + (y0 + tx)].x;
        tile[xl][tx] = expf(v);
    }
    __syncthreads();
#pragma unroll
    for (int q = 0; q < 4; ++q) {
        int yl = ty + 8 * q;
        float v = tile[tx][yl];
        int o = (y0 + yl) * UW + (x0 + tx);
        scat[o] = v;
        smap[o] = 20.0f * log10f(v + 1.0f);
    }
}

// ---------------------------------------------------------------------------
// Kernel 3: fused separable conv via V_WMMA_F32_16X16X4_F32.
// Tile staged to LDS with GLOBAL_LOAD_ASYNC_TO_LDS (ASYNCcnt-tracked).
// Stage 1 (lateral, 61 taps): out16x16 = In(16x76) * Toeplitz(latw)(76x16)
// Stage 2 (axial, 23 taps):   out16x16 = Toeplitz(axw)(16x40) * Lat(40x16)
// ---------------------------------------------------------------------------
__global__ void __launch_bounds__(256)
us_conv_wmma(const float* __restrict__ scat,
             const float* __restrict__ wts,
             float* __restrict__ env,
             float* __restrict__ bmode) {
    __shared__ float s_in[IN_ROWS * IN_STRIDE];   // 48000 B
    __shared__ float s_lat[IN_ROWS * LAT_STRIDE]; // 24960 B
    __shared__ float s_axw[AX_TAPS];
    __shared__ float s_latw[LAT_TAPS];

    const int tid = threadIdx.x;
    const int y0 = blockIdx.y * OY;
    const int x0 = blockIdx.x * OX;

    if (tid < AX_TAPS)  s_axw[tid]  = wts[tid];
    if (tid < LAT_TAPS) s_latw[tid] = wts[32 + tid];

    // Async-stage reflect-padded input tile: rows y0-11..y0+84, cols x0-30..x0+93.
    // global_load_async_to_lds_b32: VDST = LDS byte address (low 32 bits of the
    // flat shared pointer = wave-relative LDS offset), VADDR = 64-bit address.
    for (int idx = tid; idx < IN_ROWS * IN_COLS; idx += 256) {
        int i = idx / IN_COLS, j = idx - i * IN_COLS;
        int gr = y0 - AX_PAD + i;
        gr = gr < 0 ? -gr : (gr >= UH ? 2 * UH - 2 - gr : gr);
        int gc = x0 - LAT_PAD + j;
        gc = gc < 0 ? -gc : (gc >= UW ? 2 * UW - 2 - gc : gc);
        unsigned lds_addr = (unsigned)(uintptr_t)(&s_in[i * IN_STRIDE + j]);
        const float* gptr = scat + (gr * UW + gc);
        asm volatile("global_load_async_to_lds_b32 %0, %1, off"
                     :: "v"(lds_addr), "v"(gptr) : "memory");
    }
    asm volatile("s_wait_asynccnt 0x0" ::: "memory");
    __syncthreads();

    const int wv   = tid >> 5;        // wave id 0..7 (wave32)
    const int lane = tid & 31;
    const int m  = lane & 15;         // A: M row / B: N col
    const int hi = lane >> 4;         // lane half selects K pair

    // Tile-invariant Toeplitz operands, hoisted into VGPRs once per wave.
    v2f b1[19];                       // stage-1 B: latw band
#pragma unroll
    for (int kc = 0; kc < 19; ++kc) {
        int ka = kc * 4 + hi * 2;
        int i0 = ka - m, i1 = i0 + 1;
        b1[kc].x = ((unsigned)i0 < (unsigned)LAT_TAPS) ? s_latw[i0] : 0.0f;
        b1[kc].y = ((unsigned)i1 < (unsigned)LAT_TAPS) ? s_latw[i1] : 0.0f;
    }
    v2f a2[10];                       // stage-2 A: axw band
#pragma unroll
    for (int kc = 0; kc < 10; ++kc) {
        int ka = kc * 4 + hi * 2;
        int i0 = ka - m, i1 = i0 + 1;
        a2[kc].x = ((unsigned)i0 < (unsigned)AX_TAPS) ? s_axw[i0] : 0.0f;
        a2[kc].y = ((unsigned)i1 < (unsigned)AX_TAPS) ? s_axw[i1] : 0.0f;
    }

    // ---- stage 1: lateral conv, 24 tiles (6 row-groups x 4 col-groups) ----
    for (int t = wv; t < 24; t += 8) {
        int rg = t >> 2;              // row group (16 rows of s_in)
        int g  = t & 3;               // col group (16 output cols)
        v8f c = {0.f, 0.f, 0.f, 0.f, 0.f, 0.f, 0.f, 0.f};
        int abase = (rg * 16 + m) * IN_STRIDE + g * 16 + hi * 2;
#pragma unroll
        for (int kc = 0; kc < 19; ++kc) {
            v2f a;
            a.x = s_in[abase + kc * 4];
            a.y = s_in[abase + kc * 4 + 1];
            c = __builtin_amdgcn_wmma_f32_16x16x4_f32(
                    false, a, false, b1[kc], (short)0, c, false, false);
        }
        int ccol = g * 16 + m;
#pragma unroll
        for (int v = 0; v < 8; ++v) {
            int crow = rg * 16 + v + hi * 8;
            s_lat[crow * LAT_STRIDE + ccol] = c[v];
        }
    }
    __syncthreads();

    // ---- stage 2: axial conv, 16 tiles (4 row-groups x 4 col-groups) ----
    for (int t = wv; t < 16; t += 8) {
        int ty = t >> 2;              // output row group
        int gx = t & 3;               // output col group
        v8f c = {0.f, 0.f, 0.f, 0.f, 0.f, 0.f, 0.f, 0.f};
        int bbase = (ty * 16 + hi * 2) * LAT_STRIDE + gx * 16 + m;
#pragma unroll
        for (int kc = 0; kc < 10; ++kc) {
            v2f b;
            b.x = s_lat[bbase + kc * 4 * LAT_STRIDE];
            b.y = s_lat[bbase + (kc * 4 + 1) * LAT_STRIDE];
            c = __builtin_amdgcn_wmma_f32_16x16x4_f32(
                    false, a2[kc], false, b, (short)0, c, false, false);
        }
#pragma unroll
        for (int v = 0; v < 8; ++v) {
            int gy = y0 + ty * 16 + v + hi * 8;
            int gxx = x0 + gx * 16 + m;
            float e = c[v];
            int o = gy * UW + gxx;
            env[o] = e;
            bmode[o] = 20.0f * log10f(e + 1.0f);
        }
    }
}

extern "C" void kernel_launch(void* const* d_in, const int* in_sizes, int n_in,
                              void* d_out, int out_size, void* d_ws, size_t ws_size,
                              hipStream_t stream) {
    (void)in_sizes; (void)n_in; (void)out_size; (void)ws_size;
    const float4* raw4    = (const float4*)d_in[0];
    const float* rand_ax  = (const float*)d_in[1];
    const float* rand_lat = (const float*)d_in[2];

    float* out   = (float*)d_out;
    float* bmode = out;                                 // [1,1,H,W]
    float* smap  = out + (size_t)UH * UW;               // [1,1,H,W]
    float* env   = out + 2 * (size_t)UH * UW;           // [1,1,H,W]

    float* wsf  = (float*)d_ws;
    float* wts  = wsf;            // 128 floats of taps
    float* scat = wsf + 128;      // H*W scatterer

    us_psf_weights<<<1, 32, 0, stream>>>(rand_ax, rand_lat, wts);

    dim3 bT(32, 8), gT(UW / 32, UH / 32);
    us_transpose_exp<<<gT, bT, 0, stream>>>(raw4, scat, smap);

    dim3 gC(UW / OX, UH / OY);
    us_conv_wmma<<<gC, 256, 0, stream>>>(scat, wts, env, bmode);
}